// MambaBlock_2336462209414
// MI455X (gfx1250) — compile-verified
//
#include <hip/hip_runtime.h>
#include <cstdint>

// ---------------------------------------------------------------------------
// MambaBlock (S4D bidirectional) for gfx1250 / MI455X.
// Precision: fp32 end-to-end (pipeline is HBM-bound; fp32 WMMA is free).
// ---------------------------------------------------------------------------

#define BB    8
#define DIMC  128
#define LL    8192
#define HH    256
#define H2    512
#define NN    32

typedef unsigned int u32;
typedef float v2f __attribute__((ext_vector_type(2)));
typedef float v8f __attribute__((ext_vector_type(8)));
typedef u32   v4u __attribute__((ext_vector_type(4)));
typedef int   v8i __attribute__((ext_vector_type(8)));
typedef int   v4i __attribute__((ext_vector_type(4)));

// ---------------------------------------------------------------------------
// TDM: tensor_load_to_lds of a 2D fp32 tile (row-major tensor in global).
// D# group0/group1 built per CDNA5 ISA ch.8.3/8.4. Optional LDS row padding
// via pad_interval/pad_amount so the A-tile gets a 130-dword row pitch.
// ---------------------------------------------------------------------------
__device__ __forceinline__ void tdm_load_2d_f32(
    u32 lds_byte_off, const void* global_tile,
    u32 tile_d0, u32 tile_d1,
    u32 tensor_d0, u32 tensor_d1, u32 d0_stride_elems,
    bool pad_en, u32 pad_interval_code, u32 pad_amount_code)
{
  unsigned long long ga = (unsigned long long)(uintptr_t)global_tile;
  v4u g0;
  g0.x = 1u;                                              // count=1, user mode
  g0.y = lds_byte_off;                                    // lds_addr
  g0.z = (u32)(ga & 0xffffffffull);                       // global_addr[31:0]
  g0.w = (u32)((ga >> 32) & 0x01ffffffull) | (2u << 30);  // addr[56:32] | type=2
  u32 flags = (2u << 16);                                 // data_size = 4 bytes
  if (pad_en)
    flags |= (1u << 20) | ((pad_interval_code & 7u) << 22)
           | ((pad_amount_code & 127u) << 25);
  v8i g1;
  g1[0] = (int)flags;                                     // mask=0 (no cluster)
  g1[1] = (int)((tensor_d0 & 0xffffu) << 16);             // dim0[15:0] @ bit48
  g1[2] = (int)(((tensor_d0 >> 16) & 0xffffu) | ((tensor_d1 & 0xffffu) << 16));
  g1[3] = (int)(((tensor_d1 >> 16) & 0xffffu) | ((tile_d0 & 0xffffu) << 16));
  g1[4] = (int)(tile_d1 & 0xffffu);                       // tile_dim2 = 0
  g1[5] = (int)d0_stride_elems;                           // dim0_stride[31:0]
  g1[6] = 0;                                              // stride hi / dim1_stride
  g1[7] = 0;
  v4i z4 = {0, 0, 0, 0};
#if defined(__clang_major__) && (__clang_major__ >= 23)
  v8i z8 = {0, 0, 0, 0, 0, 0, 0, 0};
  __builtin_amdgcn_tensor_load_to_lds(g0, g1, z4, z4, z8, 0);
#else
  __builtin_amdgcn_tensor_load_to_lds(g0, g1, z4, z4, 0);
#endif
}

__device__ __forceinline__ float wave_sum32(float v) {
#pragma unroll
  for (int m = 16; m >= 1; m >>= 1) v += __shfl_xor(v, m, 32);
  return v;
}

// ---------------------------------------------------------------------------
// Kernel 1: discretization.  lambda = exp(dt*A);  Cd = 2*(C)*(exp(dtA)-1)/A
// (the 2x of "2*real(...)" is folded into Cd).
// ---------------------------------------------------------------------------
__global__ __launch_bounds__(256)
void s4d_param_kernel(const float* __restrict__ log_dt,
                      const float* __restrict__ A_re_log,
                      const float* __restrict__ A_im,
                      const float* __restrict__ C_re,
                      const float* __restrict__ C_im,
                      float* __restrict__ lam_re, float* __restrict__ lam_im,
                      float* __restrict__ cd0r, float* __restrict__ cd0i,
                      float* __restrict__ cd1r, float* __restrict__ cd1i)
{
  int idx = blockIdx.x * blockDim.x + threadIdx.x;
  if (idx >= HH * NN) return;
  int h = idx / NN;
  float dt = expf(log_dt[h]);
  float ar = -expf(A_re_log[idx]);
  float ai = A_im[idx];
  float dr = dt * ar, di = dt * ai;
  float er = expf(dr);
  float lr = er * cosf(di), li = er * sinf(di);
  lam_re[idx] = lr;
  lam_im[idx] = li;
  float nr = lr - 1.0f, ni = li;                 // exp(dtA) - 1
  float den = ar * ar + ai * ai;
  float fr = (nr * ar + ni * ai) / den;          // (exp(dtA)-1)/A
  float fi = (ni * ar - nr * ai) / den;
#pragma unroll
  for (int d = 0; d < 2; ++d) {
    float cr = C_re[d * HH * NN + idx], ci = C_im[d * HH * NN + idx];
    float rr = 2.0f * (cr * fr - ci * fi);
    float ii = 2.0f * (cr * fi + ci * fr);
    if (d == 0) { cd0r[idx] = rr; cd0i[idx] = ii; }
    else        { cd1r[idx] = rr; cd1i[idx] = ii; }
  }
}

// ---------------------------------------------------------------------------
// Kernel 2: proj_in  z = w_in @ x + b_in  -> gate (rows 0..255), h (256..511)
// Block: 256 thr = 8 waves, tile M=64 x N=32, K=128 via 32x WMMA f32 16x16x4.
// A/B tiles staged in LDS by the Tensor Data Mover (A with 130-dword pitch).
// ---------------------------------------------------------------------------
__global__ __launch_bounds__(256)
void gemm_in_kernel(const float* __restrict__ x, const float* __restrict__ w_in,
                    const float* __restrict__ b_in,
                    float* __restrict__ gbuf, float* __restrict__ hbuf)
{
  const int l0 = blockIdx.x * 32;
  const int m0 = blockIdx.y * 64;
  const int b  = blockIdx.z;
  __shared__ float sw[64 * 130];
  __shared__ float sx[128 * 32];
  const int tid = threadIdx.x;

  if (tid < 32) {  // wave 0 drives the TDM; TENSORcnt is per-wave
    tdm_load_2d_f32((u32)(uintptr_t)&sw[0], w_in + (size_t)m0 * DIMC,
                    /*tile*/ DIMC, 64, /*tensor*/ DIMC, H2, /*stride*/ DIMC,
                    /*pad: every 128 dw insert 2 dw*/ true, 6, 1);
    tdm_load_2d_f32((u32)(uintptr_t)&sx[0],
                    x + ((size_t)b * DIMC) * LL + l0,
                    32, DIMC, LL, DIMC, LL, false, 0, 0);
    __builtin_amdgcn_s_wait_tensorcnt(0);
  }
  __syncthreads();

  const int lane = tid & 31, l16 = lane & 15, lhi = lane >> 4;
  const int wave = tid >> 5,  wm  = wave >> 1, wn  = wave & 1;
  const float* swrow = &sw[(wm * 16 + l16) * 130];
  v8f acc = {0.f, 0.f, 0.f, 0.f, 0.f, 0.f, 0.f, 0.f};
#pragma unroll
  for (int k = 0; k < DIMC; k += 4) {
    const int kr = k + 2 * lhi;                      // ISA f32 A/B lane layout
    v2f a;  a[0]  = swrow[kr];             a[1]  = swrow[kr + 1];
    v2f bb; bb[0] = sx[kr * 32 + wn * 16 + l16];
            bb[1] = sx[(kr + 1) * 32 + wn * 16 + l16];
    acc = __builtin_amdgcn_wmma_f32_16x16x4_f32(false, a, false, bb,
                                                (short)0, acc, false, false);
  }
  const int col = l0 + wn * 16 + l16;
#pragma unroll
  for (int r = 0; r < 8; ++r) {
    const int row = m0 + wm * 16 + r + 8 * lhi;      // C: vgpr r -> M=r / r+8
    const float v = acc[r] + b_in[row];
    if (row < HH) gbuf[((size_t)b * HH + row) * LL + col] = v;
    else          hbuf[((size_t)b * HH + (row - HH)) * LL + col] = v;
  }
}

// ---------------------------------------------------------------------------
// Kernel 3: bidirectional S4D scan (exact ZOH recurrence == the FFT conv),
// fused with D-skip, SiLU gating and GroupNorm statistics.
// One wave per (b,h); lane = state index n. Chunk-of-32 register buffering
// turns the per-step scalar result into one coalesced b32 store per lane.
// ---------------------------------------------------------------------------
__global__ __launch_bounds__(256)
void ssm_scan_kernel(const float* __restrict__ hbuf, const float* __restrict__ gbuf,
                     const float* __restrict__ lam_re, const float* __restrict__ lam_im,
                     const float* __restrict__ cd0r, const float* __restrict__ cd0i,
                     const float* __restrict__ cd1r, const float* __restrict__ cd1i,
                     const float* __restrict__ Dp,
                     float* __restrict__ ybuf, float* __restrict__ stats)
{
  const int wid  = blockIdx.x * (blockDim.x >> 5) + (threadIdx.x >> 5);
  const int lane = threadIdx.x & 31;
  const int b = wid / HH, hc = wid % HH;
  const int pidx = hc * NN + lane;
  const float lr = lam_re[pidx], li = lam_im[pidx];
  const float c0r = cd0r[pidx], c0i = cd0i[pidx];
  const float c1r = cd1r[pidx], c1i = cd1i[pidx];
  const float dD = Dp[hc];
  const float* up = hbuf + ((size_t)b * HH + hc) * LL;
  const float* gp = gbuf + ((size_t)b * HH + hc) * LL;
  float*       yp = ybuf + ((size_t)b * HH + hc) * LL;

  // forward (causal) direction
  float sr = 0.f, si = 0.f;
  for (int t0 = 0; t0 < LL; t0 += 32) {
    float yreg = 0.f;
    for (int j = 0; j < 32; ++j) {
      float u   = up[t0 + j];
      float nsr = fmaf(lr, sr, fmaf(-li, si, u));   // s = lam*s + u
      float nsi = fmaf(lr, si, li * sr);
      sr = nsr; si = nsi;
      float c = fmaf(c0r, sr, -(c0i * si));         // Re(Cd0 * s)
      c = wave_sum32(c);
      if (lane == j) yreg = c;
    }
    yp[t0 + lane] = yreg;
  }

  // backward (anti-causal) direction + D skip + SiLU gate + GN stats
  sr = 0.f; si = 0.f;
  float lsum = 0.f, lsq = 0.f;
  for (int t0 = LL - 32; t0 >= 0; t0 -= 32) {
    float acc = 0.f;
    for (int j = 31; j >= 0; --j) {
      float u   = up[t0 + j];
      float nsr = fmaf(lr, sr, fmaf(-li, si, u));
      float nsi = fmaf(lr, si, li * sr);
      sr = nsr; si = nsi;
      float c = fmaf(c1r, sr, -(c1i * si));         // Re(Cd1 * s)
      c = wave_sum32(c);
      if (lane == j) acc = c;
    }
    float u  = up[t0 + lane];
    float yt = yp[t0 + lane] + acc + dD * u;
    float g  = gp[t0 + lane];
    float yg = (g / (1.f + expf(-g))) * yt;         // silu(gate) * y
    yp[t0 + lane] = yg;
    lsum += yg; lsq += yg * yg;
  }
  lsum = wave_sum32(lsum);
  lsq  = wave_sum32(lsq);
  if (lane == 0) {
    atomicAdd(&stats[2 * b],     lsum);
    atomicAdd(&stats[2 * b + 1], lsq);
  }
}

// ---------------------------------------------------------------------------
// Kernel 4: fold GroupNorm into a per-(b,h) affine:  yn = y*sc + sh
// ---------------------------------------------------------------------------
__global__ __launch_bounds__(256)
void gn_affine_kernel(const float* __restrict__ stats,
                      const float* __restrict__ gn_w, const float* __restrict__ gn_b,
                      float* __restrict__ scb, float* __restrict__ shb)
{
  int idx = blockIdx.x * blockDim.x + threadIdx.x;
  if (idx >= BB * HH) return;
  int b = idx / HH, h = idx % HH;
  float cnt = (float)HH * (float)LL;
  float mu  = stats[2 * b] / cnt;
  float var = stats[2 * b + 1] / cnt - mu * mu;
  float inv = rsqrtf(var + 1e-5f);
  scb[idx] = inv * gn_w[h];
  shb[idx] = gn_b[h] - mu * inv * gn_w[h];
}

// ---------------------------------------------------------------------------
// Kernel 5: proj_out  out = w_out @ GN(y) + b_out.  GN affine fused into the
// B-operand load; K=256 streamed in two TDM-staged chunks of 128.
// ---------------------------------------------------------------------------
__global__ __launch_bounds__(256)
void gemm_out_kernel(const float* __restrict__ ybuf, const float* __restrict__ w_out,
                     const float* __restrict__ b_out,
                     const float* __restrict__ scb, const float* __restrict__ shb,
                     float* __restrict__ out)
{
  const int l0 = blockIdx.x * 32;
  const int m0 = blockIdx.y * 64;
  const int b  = blockIdx.z;
  __shared__ float sw[64 * 130];
  __shared__ float sy[128 * 32];
  __shared__ float sc_s[HH], sh_s[HH];
  const int tid = threadIdx.x;
  for (int i = tid; i < HH; i += 256) {
    sc_s[i] = scb[b * HH + i];
    sh_s[i] = shb[b * HH + i];
  }

  const int lane = tid & 31, l16 = lane & 15, lhi = lane >> 4;
  const int wave = tid >> 5,  wm  = wave >> 1, wn  = wave & 1;
  v8f acc = {0.f, 0.f, 0.f, 0.f, 0.f, 0.f, 0.f, 0.f};

  for (int kc = 0; kc < HH; kc += 128) {
    __syncthreads();                       // prior reads done / sc_s ready
    if (tid < 32) {
      tdm_load_2d_f32((u32)(uintptr_t)&sw[0], w_out + (size_t)m0 * HH + kc,
                      128, 64, HH, DIMC, HH, true, 6, 1);
      tdm_load_2d_f32((u32)(uintptr_t)&sy[0],
                      ybuf + ((size_t)b * HH + kc) * LL + l0,
                      32, 128, LL, 128, LL, false, 0, 0);
      __builtin_amdgcn_s_wait_tensorcnt(0);
    }
    __syncthreads();

    const float* swrow = &sw[(wm * 16 + l16) * 130];
#pragma unroll
    for (int k = 0; k < 128; k += 4) {
      const int kr  = k + 2 * lhi;
      const int kg0 = kc + kr, kg1 = kg0 + 1;
      v2f a;  a[0]  = swrow[kr];             a[1]  = swrow[kr + 1];
      v2f bb; bb[0] = fmaf(sy[kr * 32 + wn * 16 + l16],       sc_s[kg0], sh_s[kg0]);
              bb[1] = fmaf(sy[(kr + 1) * 32 + wn * 16 + l16], sc_s[kg1], sh_s[kg1]);
      acc = __builtin_amdgcn_wmma_f32_16x16x4_f32(false, a, false, bb,
                                                  (short)0, acc, false, false);
    }
  }
  const int col = l0 + wn * 16 + l16;
#pragma unroll
  for (int r = 0; r < 8; ++r) {
    const int row = m0 + wm * 16 + r + 8 * lhi;
    out[((size_t)b * DIMC + row) * LL + col] = acc[r] + b_out[row];
  }
}

// ---------------------------------------------------------------------------
// Host launcher (graph-capture safe: only async ops on `stream`).
// ---------------------------------------------------------------------------
extern "C" void kernel_launch(void* const* d_in, const int* in_sizes, int n_in,
                              void* d_out, int out_size, void* d_ws, size_t ws_size,
                              hipStream_t stream)
{
  const float* x        = (const float*)d_in[0];
  const float* w_in     = (const float*)d_in[1];
  const float* b_in     = (const float*)d_in[2];
  const float* log_dt   = (const float*)d_in[3];
  const float* A_re_log = (const float*)d_in[4];
  const float* A_im     = (const float*)d_in[5];
  const float* C_re     = (const float*)d_in[6];
  const float* C_im     = (const float*)d_in[7];
  const float* Dp       = (const float*)d_in[8];
  const float* gn_w     = (const float*)d_in[9];
  const float* gn_b     = (const float*)d_in[10];
  const float* w_out    = (const float*)d_in[11];
  const float* b_out    = (const float*)d_in[12];
  float* out = (float*)d_out;

  float* ws = (float*)d_ws;
  const size_t NBH = (size_t)BB * HH * LL;          // 16.78M floats / buffer
  float* hbuf = ws;                                 // [B,H,L]
  float* gbuf = ws + NBH;                           // [B,H,L]
  float* ybuf = ws + 2 * NBH;                       // [B,H,L]
  float* p    = ws + 3 * NBH;
  const int HN = HH * NN;
  float* lam_re = p;            float* lam_im = p + HN;
  float* cd0r = p + 2 * HN;     float* cd0i = p + 3 * HN;
  float* cd1r = p + 4 * HN;     float* cd1i = p + 5 * HN;
  float* scb  = p + 6 * HN;                          // [B,H]
  float* shb  = scb + BB * HH;                       // [B,H]
  float* stats = shb + BB * HH;                      // [B,2]

  s4d_param_kernel<<<(HN + 255) / 256, 256, 0, stream>>>(
      log_dt, A_re_log, A_im, C_re, C_im,
      lam_re, lam_im, cd0r, cd0i, cd1r, cd1i);

  gemm_in_kernel<<<dim3(LL / 32, H2 / 64, BB), 256, 0, stream>>>(
      x, w_in, b_in, gbuf, hbuf);

  hipMemsetAsync(stats, 0, 2 * BB * sizeof(float), stream);

  ssm_scan_kernel<<<dim3((BB * HH) / 8), 256, 0, stream>>>(
      hbuf, gbuf, lam_re, lam_im, cd0r, cd0i, cd1r, cd1i, Dp, ybuf, stats);

  gn_affine_kernel<<<(BB * HH + 255) / 256, 256, 0, stream>>>(
      stats, gn_w, gn_b, scb, shb);

  gemm_out_kernel<<<dim3(LL / 32, DIMC / 64, BB), 256, 0, stream>>>(
      ybuf, w_out, b_out, scb, shb, out);
}